// GAT_5257039970464
// MI455X (gfx1250) — compile-verified
//
#include <hip/hip_runtime.h>
#include <math.h>

#define N_NODES   50000
#define N_FEAT    512
#define HIDDEN    128
#define N_CLASSES 7
#define N_EDGES   1600000
#define ET        (N_EDGES + N_NODES)   // edges + self loops
#define NEG_SLOPE 0.2f

typedef __attribute__((ext_vector_type(2))) float v2f;
typedef __attribute__((ext_vector_type(8))) float v8f;

// ---------------------------------------------------------------------------
// GEMM: H[N,128] = X[N,K] @ W[K,128] using V_WMMA_F32_16X16X4_F32.
// One block = 8 waves = one 16-row strip; each wave owns one 16-col tile.
// 16xK fp32 strip of X staged in LDS (contiguous in memory), shared by waves.
// ---------------------------------------------------------------------------
template <int K>
__global__ __launch_bounds__(256) void gat_gemm_wmma(const float* __restrict__ X,
                                                     const float* __restrict__ W,
                                                     float* __restrict__ H) {
    __shared__ float lds[16 * K];
    const int strip = blockIdx.x;            // row strip: rows [16*strip, 16*strip+16)
    const int tid   = threadIdx.x;

    const float* xrow = X + (size_t)strip * 16 * K;
    for (int i = tid * 4; i < 16 * K; i += 256 * 4)
        *(float4*)(lds + i) = *(const float4*)(xrow + i);
    __syncthreads();

    const int wave  = tid >> 5;              // 0..7 -> column tile
    const int lane  = tid & 31;
    const int m     = lane & 15;             // A row within tile / B col within tile
    const int khalf = (lane >> 4) * 2;       // K-pair select: 0 or 2
    const int col   = wave * 16 + m;         // global output column

    v8f acc = {};
    for (int kk = 0; kk < K; kk += 4) {
        const int kb = kk + khalf;
        // A fragment (16x4 f32 layout): lane holds A[m][kb], A[m][kb+1]
        v2f a = *(const v2f*)(lds + m * K + kb);
        // B fragment (4x16 f32 layout): lane holds W[kb][col], W[kb+1][col]
        v2f b;
        b.x = W[(size_t)kb * HIDDEN + col];
        b.y = W[(size_t)(kb + 1) * HIDDEN + col];
        acc = __builtin_amdgcn_wmma_f32_16x16x4_f32(false, a, false, b,
                                                    (short)0, acc, false, false);
    }

    // D layout: VGPR r -> row (r + 8*(lane>>4)), col (lane&15) within tile
    float* hbase    = H + ((size_t)strip * 16) * HIDDEN + wave * 16;
    const int nsel  = lane & 15;
    const int mbase = (lane >> 4) * 8;
#pragma unroll
    for (int r = 0; r < 8; ++r)
        hbase[(size_t)(mbase + r) * HIDDEN + nsel] = acc[r];
}

// ---------------------------------------------------------------------------
// Per-node attention scores: a_src[n] = h[n].att_s ; a_dst[n] = h[n].att_d
// One wave32 per node, float4 per lane, shuffle reduction.
// ---------------------------------------------------------------------------
__global__ __launch_bounds__(256) void gat_attn(const float* __restrict__ H,
                                                const float* __restrict__ att_s,
                                                const float* __restrict__ att_d,
                                                float* __restrict__ a_src,
                                                float* __restrict__ a_dst) {
    const int node = (blockIdx.x * blockDim.x + threadIdx.x) >> 5;
    const int lane = threadIdx.x & 31;
    if (node >= N_NODES) return;
    float4 hv = *(const float4*)(H + (size_t)node * HIDDEN + lane * 4);
    float4 s4 = *(const float4*)(att_s + lane * 4);
    float4 d4 = *(const float4*)(att_d + lane * 4);
    float ps = hv.x * s4.x + hv.y * s4.y + hv.z * s4.z + hv.w * s4.w;
    float pd = hv.x * d4.x + hv.y * d4.y + hv.z * d4.z + hv.w * d4.w;
    for (int off = 16; off > 0; off >>= 1) {
        ps += __shfl_down(ps, off, 32);
        pd += __shfl_down(pd, off, 32);
    }
    if (lane == 0) { a_src[node] = ps; a_dst[node] = pd; }
}

// ---------------------------------------------------------------------------
// Edge helpers
// ---------------------------------------------------------------------------
__device__ __forceinline__ void edge_sd(int e, const int* __restrict__ ei, int& s, int& d) {
    if (e < N_EDGES) { s = ei[e]; d = ei[N_EDGES + e]; }
    else             { s = d = e - N_EDGES; }          // self loop
}

__device__ __forceinline__ float leaky(float v) {
    return v > 0.f ? v : NEG_SLOPE * v;
}

__device__ __forceinline__ void atomicMaxF32(float* addr, float val) {
    int* ia  = (int*)addr;
    int  old = __hip_atomic_load(ia, __ATOMIC_RELAXED, __HIP_MEMORY_SCOPE_AGENT);
    while (__int_as_float(old) < val) {
        int assumed = old;
        old = atomicCAS(ia, assumed, __float_as_int(val));
        if (old == assumed) break;
    }
}

__global__ void gat_init(float* emax, float* denom, float* agg) {
    const int i = blockIdx.x * blockDim.x + threadIdx.x;
    if (i < N_NODES) { emax[i] = -__builtin_inff(); denom[i] = 0.f; }
    if (agg && i < N_NODES * HIDDEN) agg[i] = 0.f;
}

__global__ void gat_edge_max(const int* __restrict__ ei,
                             const float* __restrict__ a_src,
                             const float* __restrict__ a_dst,
                             float* __restrict__ emax) {
    const int e = blockIdx.x * blockDim.x + threadIdx.x;
    if (e >= ET) return;
    int s, d; edge_sd(e, ei, s, d);
    atomicMaxF32(&emax[d], leaky(a_src[s] + a_dst[d]));
}

__global__ void gat_edge_alpha(const int* __restrict__ ei,
                               const float* __restrict__ a_src,
                               const float* __restrict__ a_dst,
                               const float* __restrict__ emax,
                               float* __restrict__ alpha,
                               float* __restrict__ denom) {
    const int e = blockIdx.x * blockDim.x + threadIdx.x;
    if (e >= ET) return;
    int s, d; edge_sd(e, ei, s, d);
    float a = __expf(leaky(a_src[s] + a_dst[d]) - emax[d]);
    alpha[e] = a;
    atomicAdd(&denom[d], a);
}

// One wave32 per edge: lanes carry float4 slices of the 128-wide feature row.
// h (25.6 MB) and out (25.6 MB) both live in the 192 MB L2 -> L2-local traffic.
__global__ __launch_bounds__(256) void gat_edge_aggr(const int* __restrict__ ei,
                                                     const float* __restrict__ alpha,
                                                     const float* __restrict__ denom,
                                                     const float* __restrict__ H,
                                                     float* __restrict__ out) {
    const int e    = (blockIdx.x * blockDim.x + threadIdx.x) >> 5;
    const int lane = threadIdx.x & 31;
    if (e >= ET) return;
    int s, d; edge_sd(e, ei, s, d);
    const float w = alpha[e] / denom[d];
    float4 hv = *(const float4*)(H + (size_t)s * HIDDEN + lane * 4);
    float* o  = out + (size_t)d * HIDDEN + lane * 4;
    atomicAdd(o + 0, hv.x * w);
    atomicAdd(o + 1, hv.y * w);
    atomicAdd(o + 2, hv.z * w);
    atomicAdd(o + 3, hv.w * w);
}

__global__ void gat_bias_relu(float* __restrict__ A, const float* __restrict__ b) {
    const int i = blockIdx.x * blockDim.x + threadIdx.x;
    if (i >= N_NODES * HIDDEN) return;
    float v = A[i] + b[i & (HIDDEN - 1)];
    A[i] = v > 0.f ? v : 0.f;
}

// ---------------------------------------------------------------------------
// Output head: logits = h @ Wout + bout (128x7), softmax over 7 classes.
// One wave32 per node; 7 shuffle reductions.
// ---------------------------------------------------------------------------
__global__ __launch_bounds__(256) void gat_out(const float* __restrict__ H,
                                               const float* __restrict__ Wout,
                                               const float* __restrict__ bout,
                                               float* __restrict__ out) {
    const int node = (blockIdx.x * blockDim.x + threadIdx.x) >> 5;
    const int lane = threadIdx.x & 31;
    if (node >= N_NODES) return;
    float4 hv = *(const float4*)(H + (size_t)node * HIDDEN + lane * 4);
    float s[N_CLASSES];
#pragma unroll
    for (int c = 0; c < N_CLASSES; ++c) {
        s[c] = hv.x * Wout[(lane * 4 + 0) * N_CLASSES + c]
             + hv.y * Wout[(lane * 4 + 1) * N_CLASSES + c]
             + hv.z * Wout[(lane * 4 + 2) * N_CLASSES + c]
             + hv.w * Wout[(lane * 4 + 3) * N_CLASSES + c];
    }
#pragma unroll
    for (int c = 0; c < N_CLASSES; ++c)
        for (int off = 16; off > 0; off >>= 1)
            s[c] += __shfl_down(s[c], off, 32);
    if (lane == 0) {
        float m = -__builtin_inff();
#pragma unroll
        for (int c = 0; c < N_CLASSES; ++c) { s[c] += bout[c]; m = fmaxf(m, s[c]); }
        float sum = 0.f;
#pragma unroll
        for (int c = 0; c < N_CLASSES; ++c) { s[c] = __expf(s[c] - m); sum += s[c]; }
        const float inv = 1.f / sum;
#pragma unroll
        for (int c = 0; c < N_CLASSES; ++c) out[(size_t)node * N_CLASSES + c] = s[c] * inv;
    }
}

// ---------------------------------------------------------------------------
// Host launcher
// ---------------------------------------------------------------------------
extern "C" void kernel_launch(void* const* d_in, const int* in_sizes, int n_in,
                              void* d_out, int out_size, void* d_ws, size_t ws_size,
                              hipStream_t stream) {
    (void)in_sizes; (void)n_in; (void)out_size; (void)ws_size;

    const float* x        = (const float*)d_in[0];
    const int*   ei       = (const int*)  d_in[1];   // [2, N_EDGES] row-major
    const float* W1       = (const float*)d_in[2];
    const float* att_src1 = (const float*)d_in[3];
    const float* att_dst1 = (const float*)d_in[4];
    const float* b1       = (const float*)d_in[5];
    const float* W2       = (const float*)d_in[6];
    const float* att_src2 = (const float*)d_in[7];
    const float* att_dst2 = (const float*)d_in[8];
    const float* b2       = (const float*)d_in[9];
    const float* Wout     = (const float*)d_in[10];
    const float* bout     = (const float*)d_in[11];
    float*       out      = (float*)d_out;

    // Workspace carve-up (~59 MB of fp32)
    float* ws    = (float*)d_ws;
    float* hbuf  = ws;                                   // [N, 128]
    float* agg   = hbuf + (size_t)N_NODES * HIDDEN;      // [N, 128]
    float* asrc  = agg  + (size_t)N_NODES * HIDDEN;      // [N]
    float* adst  = asrc + N_NODES;                       // [N]
    float* emax  = adst + N_NODES;                       // [N]
    float* denom = emax + N_NODES;                       // [N]
    float* alpha = denom + N_NODES;                      // [ET]

    const int  THREADS     = 256;
    const dim3 blk(THREADS);
    const dim3 grid_gemm(N_NODES / 16);                                  // 3125
    const dim3 grid_nh((N_NODES * HIDDEN + THREADS - 1) / THREADS);
    const dim3 grid_node_wave((N_NODES * 32 + THREADS - 1) / THREADS);
    const dim3 grid_edge((ET + THREADS - 1) / THREADS);
    const dim3 grid_edge_wave(ET / 8);                                   // 206250

    // ---- Layer 1 ----
    gat_init<<<grid_nh, blk, 0, stream>>>(emax, denom, agg);
    gat_gemm_wmma<N_FEAT><<<grid_gemm, blk, 0, stream>>>(x, W1, hbuf);
    gat_attn<<<grid_node_wave, blk, 0, stream>>>(hbuf, att_src1, att_dst1, asrc, adst);
    gat_edge_max<<<grid_edge, blk, 0, stream>>>(ei, asrc, adst, emax);
    gat_edge_alpha<<<grid_edge, blk, 0, stream>>>(ei, asrc, adst, emax, alpha, denom);
    gat_edge_aggr<<<grid_edge_wave, blk, 0, stream>>>(ei, alpha, denom, hbuf, agg);
    gat_bias_relu<<<grid_nh, blk, 0, stream>>>(agg, b1);                 // agg = layer-2 input

    // ---- Layer 2 ----
    gat_gemm_wmma<HIDDEN><<<grid_gemm, blk, 0, stream>>>(agg, W2, hbuf);
    gat_init<<<grid_nh, blk, 0, stream>>>(emax, denom, agg);             // agg now accumulator
    gat_attn<<<grid_node_wave, blk, 0, stream>>>(hbuf, att_src2, att_dst2, asrc, adst);
    gat_edge_max<<<grid_edge, blk, 0, stream>>>(ei, asrc, adst, emax);
    gat_edge_alpha<<<grid_edge, blk, 0, stream>>>(ei, asrc, adst, emax, alpha, denom);
    gat_edge_aggr<<<grid_edge_wave, blk, 0, stream>>>(ei, alpha, denom, hbuf, agg);
    gat_bias_relu<<<grid_nh, blk, 0, stream>>>(agg, b2);

    // ---- Output head ----
    gat_out<<<grid_node_wave, blk, 0, stream>>>(agg, Wout, bout, out);
}